// CausalStdMeanScaler_47717086658894
// MI455X (gfx1250) — compile-verified
//
#include <hip/hip_runtime.h>
#include <hip/hip_bf16.h>
#include <stdint.h>

// Causal weighted running mean/std scaler.
//   rows = B*V = 4096, T = 8192 (fixed by the reference)
//   One wave32 per row; async global->LDS double-buffered streaming.
//   8 f32 per lane per chunk (CHUNK = 256), approx rcp/sqrt (~1 ulp).

#define T_LEN           8192
#define ELEMS_PER_LANE  8
#define CHUNK           (32 * ELEMS_PER_LANE)   // 256
#define WAVES_PER_BLOCK 8
#define NCHUNK          (T_LEN / CHUNK)         // 32
#define MINIMUM_SCALE   0.1f

typedef __attribute__((ext_vector_type(4))) float v4f;
typedef __attribute__((ext_vector_type(4))) int   v4i;

// ---- gfx1250 async global->LDS copy (ASYNCcnt-tracked DMA into LDS) ----
__device__ __forceinline__ void async_copy_b128(uint32_t lds_off, const void* gptr) {
    asm volatile("global_load_async_to_lds_b128 %0, %1, off"
                 :
                 : "v"(lds_off), "v"((unsigned long long)(uintptr_t)gptr)
                 : "memory");
}
#define WAIT_ASYNC_LE6() asm volatile("s_wait_asynccnt 0x6" ::: "memory")
#define WAIT_ASYNC_LE0() asm volatile("s_wait_asynccnt 0x0" ::: "memory")

__device__ __forceinline__ uint32_t lds_off_of(const void* p) {
    // Generic LDS pointers on amdgcn are {aperture_hi32, lds_offset32}:
    // the low 32 bits are the workgroup-relative DS address.
    return (uint32_t)(uintptr_t)p;
}

__global__ __launch_bounds__(WAVES_PER_BLOCK * 32)
void causal_scaler_kernel(const float* __restrict__ data,
                          const int*   __restrict__ mask,
                          const float* __restrict__ weights,
                          float*       __restrict__ out,
                          long long n_per_tensor,
                          long long n_rows) {
    __shared__ __align__(16) float s_d[WAVES_PER_BLOCK][2][CHUNK];
    __shared__ __align__(16) float s_w[WAVES_PER_BLOCK][2][CHUNK];
    __shared__ __align__(16) int   s_m[WAVES_PER_BLOCK][2][CHUNK];

    const int lane = (int)(threadIdx.x & 31u);
    const int wave = (int)(threadIdx.x >> 5u);
    const long long row = (long long)blockIdx.x * WAVES_PER_BLOCK + wave;
    if (row >= n_rows) return;

    const long long rowOff = row * (long long)T_LEN;
    const float* dRow = data    + rowOff;
    const int*   mRow = mask    + rowOff;
    const float* wRow = weights + rowOff;
    float* o_scaled = out                      + rowOff;
    float* o_mean   = out +     n_per_tensor   + rowOff;
    float* o_scale  = out + 2 * n_per_tensor   + rowOff;

    const int laneSlot = lane * ELEMS_PER_LANE;   // first element index of this lane
    const uint32_t ld_d[2] = { lds_off_of(&s_d[wave][0][laneSlot]),
                               lds_off_of(&s_d[wave][1][laneSlot]) };
    const uint32_t ld_w[2] = { lds_off_of(&s_w[wave][0][laneSlot]),
                               lds_off_of(&s_w[wave][1][laneSlot]) };
    const uint32_t ld_m[2] = { lds_off_of(&s_m[wave][0][laneSlot]),
                               lds_off_of(&s_m[wave][1][laneSlot]) };

    // Prologue: stage chunk 0 into buffer 0 (6 async b128 ops: 2 per array).
    {
        const int off = laneSlot;
        async_copy_b128(ld_d[0],      dRow + off);
        async_copy_b128(ld_d[0] + 16, dRow + off + 4);
        async_copy_b128(ld_w[0],      wRow + off);
        async_copy_b128(ld_w[0] + 16, wRow + off + 4);
        async_copy_b128(ld_m[0],      mRow + off);
        async_copy_b128(ld_m[0] + 16, mRow + off + 4);
    }

    float carry_w = 0.0f, carry_v = 0.0f, carry_r = 0.0f;

    for (int c = 0; c < NCHUNK; ++c) {
        const int st = c & 1;

        // Stage chunk c+1 into the other buffer, then wait for chunk c.
        if (c + 1 < NCHUNK) {
            const int off = (c + 1) * CHUNK + laneSlot;
            const int bf = st ^ 1;
            async_copy_b128(ld_d[bf],      dRow + off);
            async_copy_b128(ld_d[bf] + 16, dRow + off + 4);
            async_copy_b128(ld_w[bf],      wRow + off);
            async_copy_b128(ld_w[bf] + 16, wRow + off + 4);
            async_copy_b128(ld_m[bf],      mRow + off);
            async_copy_b128(ld_m[bf] + 16, mRow + off + 4);
            WAIT_ASYNC_LE6();   // async loads retire in order: chunk c has landed
        } else {
            WAIT_ASYNC_LE0();
        }

        // ---- consume chunk c from LDS ----
        const v4f dA = *(const v4f*)&s_d[wave][st][laneSlot];
        const v4f dB = *(const v4f*)&s_d[wave][st][laneSlot + 4];
        const v4f qA = *(const v4f*)&s_w[wave][st][laneSlot];
        const v4f qB = *(const v4f*)&s_w[wave][st][laneSlot + 4];
        const v4i mA = *(const v4i*)&s_m[wave][st][laneSlot];
        const v4i mB = *(const v4i*)&s_m[wave][st][laneSlot + 4];

        float x[ELEMS_PER_LANE], w[ELEMS_PER_LANE];
        x[0] = dA.x; x[1] = dA.y; x[2] = dA.z; x[3] = dA.w;
        x[4] = dB.x; x[5] = dB.y; x[6] = dB.z; x[7] = dB.w;
        w[0] = qA.x * (float)mA.x; w[1] = qA.y * (float)mA.y;
        w[2] = qA.z * (float)mA.z; w[3] = qA.w * (float)mA.w;
        w[4] = qB.x * (float)mB.x; w[5] = qB.y * (float)mB.y;
        w[6] = qB.z * (float)mB.z; w[7] = qB.w * (float)mB.w;

        // Per-lane inclusive prefixes of w and x*w.
        float pw[ELEMS_PER_LANE], pv[ELEMS_PER_LANE];
        float aw = 0.0f, av = 0.0f;
        #pragma unroll
        for (int i = 0; i < ELEMS_PER_LANE; ++i) {
            aw += w[i];        pw[i] = aw;
            av += x[i] * w[i]; pv[i] = av;
        }

        // Wave-level inclusive scan of lane totals (w, v fused).
        float iw = aw, iv = av;
        #pragma unroll
        for (int dlt = 1; dlt < 32; dlt <<= 1) {
            const float sa = __shfl_up(iw, (unsigned)dlt, 32);
            const float sb = __shfl_up(iv, (unsigned)dlt, 32);
            if (lane >= dlt) { iw += sa; iv += sb; }
        }
        const float exw = carry_w + (iw - aw);   // exclusive-prefix + carry
        const float exv = carry_v + (iv - av);

        // Running means + residual terms (rcp ~1 ulp; den >= 1 so it's safe).
        float rden[ELEMS_PER_LANE], mu[ELEMS_PER_LANE], e[ELEMS_PER_LANE];
        float pr[ELEMS_PER_LANE];
        float ar = 0.0f;
        #pragma unroll
        for (int i = 0; i < ELEMS_PER_LANE; ++i) {
            const float den = fmaxf(exw + pw[i], 1.0f);
            const float rd  = __builtin_amdgcn_rcpf(den);
            rden[i] = rd;
            const float m = (exv + pv[i]) * rd;
            mu[i] = m;
            const float r = x[i] - m;
            e[i] = r;
            ar += r * r * w[i];
            pr[i] = ar;
        }

        // Wave-level inclusive scan of residual totals.
        float ir = ar;
        #pragma unroll
        for (int dlt = 1; dlt < 32; dlt <<= 1) {
            const float sr = __shfl_up(ir, (unsigned)dlt, 32);
            if (lane >= dlt) ir += sr;
        }
        const float exr = carry_r + (ir - ar);

        float sc[ELEMS_PER_LANE], so[ELEMS_PER_LANE];
        #pragma unroll
        for (int i = 0; i < ELEMS_PER_LANE; ++i) {
            const float var = (exr + pr[i]) * rden[i];
            const float s   = __builtin_amdgcn_sqrtf(var + MINIMUM_SCALE);
            sc[i] = s;
            so[i] = e[i] * __builtin_amdgcn_rcpf(s);
        }

        // Carry = lane-31 inclusive totals (exclusive + lane total there).
        carry_w = __shfl(exw + aw, 31, 32);
        carry_v = __shfl(exv + av, 31, 32);
        carry_r = __shfl(exr + ar, 31, 32);

        // ---- streaming (non-temporal) outputs ----
        const int o = c * CHUNK + laneSlot;
        v4f t;
        t.x = so[0]; t.y = so[1]; t.z = so[2]; t.w = so[3];
        __builtin_nontemporal_store(t, (v4f*)(o_scaled + o));
        t.x = so[4]; t.y = so[5]; t.z = so[6]; t.w = so[7];
        __builtin_nontemporal_store(t, (v4f*)(o_scaled + o + 4));
        t.x = mu[0]; t.y = mu[1]; t.z = mu[2]; t.w = mu[3];
        __builtin_nontemporal_store(t, (v4f*)(o_mean + o));
        t.x = mu[4]; t.y = mu[5]; t.z = mu[6]; t.w = mu[7];
        __builtin_nontemporal_store(t, (v4f*)(o_mean + o + 4));
        t.x = sc[0]; t.y = sc[1]; t.z = sc[2]; t.w = sc[3];
        __builtin_nontemporal_store(t, (v4f*)(o_scale + o));
        t.x = sc[4]; t.y = sc[5]; t.z = sc[6]; t.w = sc[7];
        __builtin_nontemporal_store(t, (v4f*)(o_scale + o + 4));
    }
}

extern "C" void kernel_launch(void* const* d_in, const int* in_sizes, int n_in,
                              void* d_out, int out_size, void* d_ws, size_t ws_size,
                              hipStream_t stream) {
    const float* data    = (const float*)d_in[0];
    const int*   mask    = (const int*)  d_in[1];
    const float* weights = (const float*)d_in[2];
    float*       out     = (float*)d_out;

    const long long n_per_tensor = (long long)in_sizes[0];   // B*V*T
    const long long n_rows       = n_per_tensor / T_LEN;     // B*V
    const int blocks = (int)((n_rows + WAVES_PER_BLOCK - 1) / WAVES_PER_BLOCK);

    causal_scaler_kernel<<<blocks, WAVES_PER_BLOCK * 32, 0, stream>>>(
        data, mask, weights, out, n_per_tensor, n_rows);
}